// ActorCritic_28063316312889
// MI455X (gfx1250) — compile-verified
//
#include <hip/hip_runtime.h>

// ActorCritic forward, MI455X (gfx1250) — fully fused single kernel.
//
// Only the mean path feeds the outputs (the Sigma propagation in the
// reference is dead code w.r.t. (logits, value)), so the live work is:
//   h1 = relu(x @ W1 + b1)        (1024x512)*(512x256)
//   h2 = relu(h1 @ W2 + b2)       (1024x256)*(256x128)
//   logits = h2 @ Wa + ba         (1024x128)*(128x18)
//   value  = h2 @ Wc + bc         (1024x128)*(128x1)
// Every layer is batch-row independent -> one workgroup per 16 batch rows
// computes all layers with activations living in LDS (never touch HBM).
// ~3.4e8 FLOP over ~2.8MB HBM traffic: latency/bandwidth bound, so we use
// V_WMMA_F32_16X16X4_F32 (full fp32; 5e-5-scale weights would denormalize
// in f16) and minimize launch count (1 kernel).
//
// LDS pitches padded to ≡4 (mod 64 banks) so A-fragment reads
// (lane -> row, row-pitch stride) are bank-conflict free; heads weight
// panel pitch 40 so the two K-rows per half-wave use disjoint bank sets.

#define BATCH  1024
#define IN_DIM 512
#define H1D    256
#define H2D    128
#define NACT   18

// LDS layout in floats
#define X_PITCH   516            // 512 + 4
#define H1_PITCH  260            // 256 + 4
#define H2_PITCH  132            // 128 + 4
#define BP_PITCH  40             // 32 + 8 (2*40 ≡ 16 mod 64)
#define X_OFF     0              // 16 x 516 = 8256 (reused as Bpad: 128 x 40 = 5120)
#define H1_OFF    8256           // 16 x 260 = 4160
#define H2_OFF    12416          // 16 x 132 = 2112
#define LDS_FLOATS 14528         // 58112 bytes

typedef __attribute__((ext_vector_type(2))) float v2f;
typedef __attribute__((ext_vector_type(4))) float v4f;
typedef __attribute__((ext_vector_type(8))) float v8f;

__device__ __forceinline__ v8f wmma_f32_k4(v2f a, v2f b, v8f c) {
  // D(16x16,f32) = A(16x4,f32) * B(4x16,f32) + C
  return __builtin_amdgcn_wmma_f32_16x16x4_f32(
      false, a, false, b, (short)0, c, false, false);
}

__global__ __launch_bounds__(256) void actor_critic_fused(
    const float* __restrict__ x,
    const float* __restrict__ w1, const float* __restrict__ b1,
    const float* __restrict__ w2, const float* __restrict__ b2,
    const float* __restrict__ wa, const float* __restrict__ ba,
    const float* __restrict__ wc, const float* __restrict__ bc,
    float* __restrict__ out) {
  __shared__ __align__(16) float smem[LDS_FLOATS];

  const int tid  = threadIdx.x;
  const int lane = tid & 31;
  const int wave = tid >> 5;
  const int half = lane >> 4;   // selects K pair inside the 16x4 A step
  const int lidx = lane & 15;   // M row for A-frag, N col for B/C-frag
  const int m0   = blockIdx.x << 4;   // 16 batch rows per block

  // ---- Stage x tile (16 x 512) into LDS with padded pitch, b128 both sides.
  {
    const int row = tid >> 4;          // 16 threads per row
    const int cid = tid & 15;
    const v4f* __restrict__ src = (const v4f*)(x + (size_t)(m0 + row) * IN_DIM);
    v4f* dst = (v4f*)&smem[X_OFF + row * X_PITCH];
#pragma unroll
    for (int j = 0; j < 8; ++j)        // 8 * 16 = 128 float4 = 512 floats/row
      dst[cid + 16 * j] = src[cid + 16 * j];
  }
  // Pull layer-2 weights (128KB) toward L2 while layer 1 computes.
#pragma unroll
  for (int j = 0; j < 4; ++j)
    __builtin_prefetch(w2 + (size_t)(tid + 256 * j) * 32, 0, 1);
  __syncthreads();

  // ---- Layer 1: h1(16x256) = relu(xTile @ W1 + b1). 2 N-tiles per wave,
  //      shared A-fragment, B streamed from global (L2-resident).
  {
    const int n0 = wave << 5;                        // 32 output cols per wave
    const float* __restrict__ B = w1 + n0 + lidx + (size_t)(2 * half) * H1D;
    v8f acc0 = {}, acc1 = {};
#pragma unroll 4
    for (int k0 = 0; k0 < IN_DIM; k0 += 4) {
      v2f a = *(const v2f*)&smem[X_OFF + lidx * X_PITCH + 2 * half + k0];
      const size_t off = (size_t)k0 * H1D;
      v2f bA, bB;
      bA.x = B[off];       bA.y = B[off + H1D];
      bB.x = B[off + 16];  bB.y = B[off + H1D + 16];
      acc0 = wmma_f32_k4(a, bA, acc0);
      acc1 = wmma_f32_k4(a, bB, acc1);
    }
    const float bv0 = b1[n0 + lidx];
    const float bv1 = b1[n0 + 16 + lidx];
#pragma unroll
    for (int r = 0; r < 8; ++r) {
      const int m = r + 8 * half;
      smem[H1_OFF + m * H1_PITCH + n0 + lidx]      = fmaxf(acc0[r] + bv0, 0.0f);
      smem[H1_OFF + m * H1_PITCH + n0 + 16 + lidx] = fmaxf(acc1[r] + bv1, 0.0f);
    }
  }
  __syncthreads();

  // ---- Build zero-padded heads weight panel Bpad(128 x 32) over the dead
  //      x region: cols 0..17 = Wa, col 18 = Wc, 19..31 = 0.  This removes
  //      ALL masking from the heads WMMA loop (EXEC stays all-1s trivially).
  for (int i = tid; i < H2D * 32; i += 256) {
    const int k = i >> 5, c = i & 31;
    float v = 0.0f;
    if (c < NACT)       v = wa[(size_t)k * NACT + c];
    else if (c == NACT) v = wc[k];
    smem[X_OFF + k * BP_PITCH + c] = v;
  }

  // ---- Layer 2: h2(16x128) = relu(h1 @ W2 + b2). 1 N-tile per wave.
  {
    const int n0 = wave << 4;
    const float* __restrict__ B = w2 + n0 + lidx + (size_t)(2 * half) * H2D;
    v8f acc = {};
#pragma unroll 4
    for (int k0 = 0; k0 < H1D; k0 += 4) {
      v2f a = *(const v2f*)&smem[H1_OFF + lidx * H1_PITCH + 2 * half + k0];
      const size_t off = (size_t)k0 * H2D;
      v2f b;
      b.x = B[off];  b.y = B[off + H2D];
      acc = wmma_f32_k4(a, b, acc);
    }
    const float bv = b2[n0 + lidx];
#pragma unroll
    for (int r = 0; r < 8; ++r)
      smem[H2_OFF + (r + 8 * half) * H2_PITCH + n0 + lidx] =
          fmaxf(acc[r] + bv, 0.0f);
  }
  __syncthreads();

  // ---- Heads: (16x32) = h2 @ Bpad; cols 0..17 logits, col 18 value.
  //      Waves 0,1 take one 16-wide N tile each; all operands from LDS.
  if (wave < 2) {
    const int n0  = wave << 4;
    const int col = n0 + lidx;          // 0..31, valid 0..18
    v8f acc = {};
#pragma unroll 4
    for (int k0 = 0; k0 < H2D; k0 += 4) {
      const int ka = k0 + 2 * half;
      v2f a = *(const v2f*)&smem[H2_OFF + lidx * H2_PITCH + 2 * half + k0];
      v2f b;
      b.x = smem[X_OFF + ka * BP_PITCH + col];
      b.y = smem[X_OFF + (ka + 1) * BP_PITCH + col];
      acc = wmma_f32_k4(a, b, acc);
    }
    const float bva = ba[col < NACT ? col : 0];   // clamped load + select,
    const float bvc = bc[0];                      // no predicated vmem
    const float bv  = (col < NACT) ? bva : ((col == NACT) ? bvc : 0.0f);
#pragma unroll
    for (int r = 0; r < 8; ++r) {
      const int m = m0 + r + 8 * half;
      const float v = acc[r] + bv;
      if (col < NACT)
        out[(size_t)m * NACT + col] = v;           // logits (1024x18)
      else if (col == NACT)
        out[(size_t)BATCH * NACT + m] = v;         // value  (1024)
    }
  }
}

extern "C" void kernel_launch(void* const* d_in, const int* in_sizes, int n_in,
                              void* d_out, int out_size, void* d_ws, size_t ws_size,
                              hipStream_t stream) {
  (void)in_sizes; (void)n_in; (void)out_size; (void)d_ws; (void)ws_size;
  const float* x     = (const float*)d_in[0];
  const float* w_mu1 = (const float*)d_in[1];
  const float* b_mu1 = (const float*)d_in[3];
  const float* w_mu2 = (const float*)d_in[5];
  const float* b_mu2 = (const float*)d_in[7];
  const float* w_mua = (const float*)d_in[9];
  const float* b_mua = (const float*)d_in[11];
  const float* w_muc = (const float*)d_in[13];
  const float* b_muc = (const float*)d_in[15];

  // One block per 16 batch rows: 64 blocks x 256 threads (8 wave32).
  actor_critic_fused<<<BATCH / 16, 256, 0, stream>>>(
      x, w_mu1, b_mu1, w_mu2, b_mu2, w_mua, b_mua, w_muc, b_muc,
      (float*)d_out);
}